// PointNet2Encoder_35880156791485
// MI455X (gfx1250) — compile-verified
//
#include <hip/hip_runtime.h>

#define BATCH 8
#define NPTS  4096

typedef __attribute__((ext_vector_type(2))) float v2f;
typedef __attribute__((ext_vector_type(8))) float v8f;

struct LayerP {
  const float* W;   // padded weights [Cout, Kp] row-major, zero tail
  const float* bb;  // bias
  const float* g;   // gamma
  const float* be;  // beta
  const float* mu;  // running mean
  const float* var; // running var
  int Kp;           // roundup4(Cin) — padded K
  int Cout;
};

__device__ __forceinline__ int cpad(int c) { return ((c + 3) & ~3) + 4; }

// ---------------------------------------------------------------------------
// Weight padding: W [Cout, Cin] -> Wp [Cout, Kp], zero-filled K tail.
// ---------------------------------------------------------------------------
__global__ void k_pad_w(const float* __restrict__ W, float* __restrict__ Wp,
                        int Cin, int Kp, int Cout) {
  int i = blockIdx.x * blockDim.x + threadIdx.x;
  if (i >= Cout * Kp) return;
  int o = i / Kp, c = i - o * Kp;
  Wp[i] = (c < Cin) ? W[(size_t)o * Cin + c] : 0.0f;
}

// ---------------------------------------------------------------------------
// Input transpose: xyz [B,6,N] -> pts [B,N,3], feat [B,N,6]
// ---------------------------------------------------------------------------
__global__ void k_transpose_in(const float* __restrict__ xyz,
                               float* __restrict__ pts,
                               float* __restrict__ feat) {
  int i = blockIdx.x * blockDim.x + threadIdx.x;
  if (i >= BATCH * NPTS) return;
  int b = i / NPTS, n = i - b * NPTS;
#pragma unroll
  for (int c = 0; c < 6; ++c) {
    float v = xyz[(b * 6 + c) * NPTS + n];
    feat[(size_t)i * 6 + c] = v;
    if (c < 3) pts[(size_t)i * 3 + c] = v;
  }
}

// ---------------------------------------------------------------------------
// Farthest point sampling: one block per batch, dist[] lives in LDS.
// ---------------------------------------------------------------------------
__global__ __launch_bounds__(256) void k_fps(const float* __restrict__ pts,
                                             int n, int npoint,
                                             int* __restrict__ out) {
  __shared__ float dist[4096];
  __shared__ float rv[256];
  __shared__ int ri[256];
  __shared__ int curIdx;
  const int b = blockIdx.x;
  const int tid = threadIdx.x;

  for (int i = tid; i < n; i += 256) dist[i] = 1e10f;
  if (tid == 0) curIdx = 0;
  __syncthreads();

  for (int it = 0; it < npoint; ++it) {
    const int cur = curIdx;
    if (tid == 0) out[b * npoint + it] = cur;
    const float cx = pts[(size_t)(b * n + cur) * 3 + 0];
    const float cy = pts[(size_t)(b * n + cur) * 3 + 1];
    const float cz = pts[(size_t)(b * n + cur) * 3 + 2];

    float best = -1.0f;
    int bi = 0x7FFFFFFF;
    for (int i = tid; i < n; i += 256) {
      float dx = pts[(size_t)(b * n + i) * 3 + 0] - cx;
      float dy = pts[(size_t)(b * n + i) * 3 + 1] - cy;
      float dz = pts[(size_t)(b * n + i) * 3 + 2] - cz;
      float d = dx * dx + dy * dy + dz * dz;
      float nd = dist[i];
      nd = d < nd ? d : nd;
      dist[i] = nd;
      if (nd > best) { best = nd; bi = i; }
    }
    rv[tid] = best;
    ri[tid] = bi;
    __syncthreads();
    for (int off = 128; off > 0; off >>= 1) {
      if (tid < off) {
        if (rv[tid + off] > rv[tid] ||
            (rv[tid + off] == rv[tid] && ri[tid + off] < ri[tid])) {
          rv[tid] = rv[tid + off];
          ri[tid] = ri[tid + off];
        }
      }
      __syncthreads();
    }
    if (tid == 0) curIdx = ri[0];
    __syncthreads();
  }
}

// ---------------------------------------------------------------------------
// Gather centers: new_xyz[b,s,:] = pts[b, fps_idx[b,s], :]
// ---------------------------------------------------------------------------
__global__ void k_gather_centers(const float* __restrict__ pts,
                                 const int* __restrict__ idx,
                                 float* __restrict__ ctr, int n, int S) {
  int i = blockIdx.x * blockDim.x + threadIdx.x;
  if (i >= BATCH * S) return;
  int b = i / S;
  int j = idx[i];
#pragma unroll
  for (int c = 0; c < 3; ++c)
    ctr[(size_t)i * 3 + c] = pts[(size_t)(b * n + j) * 3 + c];
}

// ---------------------------------------------------------------------------
// WMMA fragment inner product over the padded K dim. Both operand streams are
// unconditionally loadable (weights zero-padded in global, activations
// zero-padded in LDS), 8B-aligned -> b64 loads, no EXEC masking. Two
// accumulator chains so loads can overlap the serial WMMA RAW chain.
// ---------------------------------------------------------------------------
__device__ __forceinline__ v8f wmma_dot(const float* __restrict__ srcRow,
                                        const float* __restrict__ Wr, int Kp,
                                        int khalf) {
  v8f acc0 = {};
  v8f acc1 = {};
  int c0 = 0;
  for (; c0 + 8 <= Kp; c0 += 8) {
    v2f a0 = *(const v2f*)(srcRow + c0 + khalf);
    v2f w0 = *(const v2f*)(Wr + c0 + khalf);
    v2f a1 = *(const v2f*)(srcRow + c0 + 4 + khalf);
    v2f w1 = *(const v2f*)(Wr + c0 + 4 + khalf);
    acc0 = __builtin_amdgcn_wmma_f32_16x16x4_f32(false, a0, false, w0,
                                                 (short)0, acc0, false, false);
    acc1 = __builtin_amdgcn_wmma_f32_16x16x4_f32(false, a1, false, w1,
                                                 (short)0, acc1, false, false);
  }
  if (c0 < Kp) {  // Kp % 8 == 4 tail
    v2f a0 = *(const v2f*)(srcRow + c0 + khalf);
    v2f w0 = *(const v2f*)(Wr + c0 + khalf);
    acc0 = __builtin_amdgcn_wmma_f32_16x16x4_f32(false, a0, false, w0,
                                                 (short)0, acc0, false, false);
  }
  return acc0 + acc1;
}

// ---------------------------------------------------------------------------
// One MLP layer over a 16-row activation tile using V_WMMA_F32_16X16X4_F32.
// ---------------------------------------------------------------------------
__device__ __forceinline__ void mlp_layer16(const float* __restrict__ src,
                                            float* __restrict__ dst,
                                            int CPin, int CPout,
                                            const LayerP& L, int tid) {
  const int lane = tid & 31;
  const int wv = tid >> 5;
  const int r = lane & 15;            // A row / B col within tile
  const int khalf = (lane >> 4) << 1; // K offset 0 or 2
  const int rb = (lane >> 4) << 3;    // C/D row base 0 or 8
  const int ntiles = L.Cout >> 4;
  const float* srcRow = src + r * CPin;
  for (int t = wv; t < ntiles; t += 8) {
    const int n0 = t << 4;
    const float* Wr = L.W + (size_t)(n0 + r) * L.Kp;
    v8f acc = wmma_dot(srcRow, Wr, L.Kp, khalf);
    const int oc = n0 + r;
    const float sc = L.g[oc] * rsqrtf(L.var[oc] + 1e-5f);
    const float bs = (L.bb[oc] - L.mu[oc]) * sc + L.be[oc];
#pragma unroll
    for (int i = 0; i < 8; ++i) {
      float v = acc[i] * sc + bs;
      dst[(rb + i) * CPout + oc] = v > 0.0f ? v : 0.0f;
    }
  }
}

// Last layer fused with max-pool: post-ReLU values are >= 0 so an integer
// atomicMax on the float bit pattern implements float max.
__device__ __forceinline__ void mlp_layer16_pool(const float* __restrict__ src,
                                                 int* __restrict__ pool,
                                                 int CPin, const LayerP& L,
                                                 int tid) {
  const int lane = tid & 31;
  const int wv = tid >> 5;
  const int r = lane & 15;
  const int khalf = (lane >> 4) << 1;
  const int ntiles = L.Cout >> 4;
  const float* srcRow = src + r * CPin;
  for (int t = wv; t < ntiles; t += 8) {
    const int n0 = t << 4;
    const float* Wr = L.W + (size_t)(n0 + r) * L.Kp;
    v8f acc = wmma_dot(srcRow, Wr, L.Kp, khalf);
    const int oc = n0 + r;
    const float sc = L.g[oc] * rsqrtf(L.var[oc] + 1e-5f);
    const float bs = (L.bb[oc] - L.mu[oc]) * sc + L.be[oc];
    float mx = 0.0f;
#pragma unroll
    for (int i = 0; i < 8; ++i) {
      float v = acc[i] * sc + bs;
      v = v > 0.0f ? v : 0.0f;
      mx = v > mx ? v : mx;
    }
    float other = __shfl_down(mx, 16, 32);  // rows 0-7 vs 8-15
    if (lane < 16) {
      mx = other > mx ? other : mx;
      atomicMax(&pool[oc], __float_as_int(mx));
    }
  }
}

// ---------------------------------------------------------------------------
// Fused set-abstraction stage: ball query (ordered ballot compaction) +
// grouping + 3-layer MLP (WMMA, LDS ping-pong, 16 rows per pass) + max-pool.
// One block (8 wave32) per (batch, center) group.
// ---------------------------------------------------------------------------
template <int CPA, int CPB>
__global__ __launch_bounds__(256) void k_sa_stage(
    const float* __restrict__ pts, const float* __restrict__ fin,
    const float* __restrict__ ctr, float* __restrict__ fout, int n, int S,
    int Cprev, float r2, LayerP L0, LayerP L1, LayerP L2) {
  __shared__ float bufA[16 * CPA];
  __shared__ float bufB[16 * CPB];
  __shared__ int nbr[32];
  __shared__ int ncnt_s;
  __shared__ int pool[512];

  const int g = blockIdx.x;
  const int b = g / S;
  const int tid = threadIdx.x;
  const int lane = tid & 31;
  const int wv = tid >> 5;

  const float cx = ctr[(size_t)g * 3 + 0];
  const float cy = ctr[(size_t)g * 3 + 1];
  const float cz = ctr[(size_t)g * 3 + 2];

  // Ordered "first 32 indices within radius" via ballot compaction (wave 0).
  if (wv == 0) {
    int cnt = 0;
    for (int base = 0; base < n && cnt < 32; base += 32) {
      int j = base + lane;
      bool inb = false;
      if (j < n) {
        float dx = pts[(size_t)(b * n + j) * 3 + 0] - cx;
        float dy = pts[(size_t)(b * n + j) * 3 + 1] - cy;
        float dz = pts[(size_t)(b * n + j) * 3 + 2] - cz;
        inb = (dx * dx + dy * dy + dz * dz) <= r2;
      }
      unsigned m = (unsigned)__ballot(inb);
      int pos = cnt + __popc(m & ((1u << lane) - 1u));
      if (inb && pos < 32) nbr[pos] = j;
      cnt += __popc(m);
    }
    if (lane == 0) {
      if (cnt == 0) { nbr[0] = 0; cnt = 1; }
      ncnt_s = cnt < 32 ? cnt : 32;
    }
  }
  for (int o = tid; o < L2.Cout; o += 256) pool[o] = 0;
  __syncthreads();

  const int cnt = ncnt_s;
  const int Cin0 = Cprev + 3;
  const int K0 = (Cin0 + 3) & ~3;  // == L0.Kp
  const int CP0 = cpad(Cin0);
  const int CP1 = cpad(L0.Cout);
  const int CP2 = cpad(L1.Cout);

  for (int half = 0; half < 2; ++half) {
    // Gather 16 rows; zero-fill K tail so the WMMA loop needs no masking.
    for (int i = tid; i < 16 * K0; i += 256) {
      int k = i / K0, c = i - k * K0;
      int kk = half * 16 + k;
      int j = nbr[kk < cnt ? kk : 0];
      float v = 0.0f;
      if (c == 0)
        v = pts[(size_t)(b * n + j) * 3 + 0] - cx;
      else if (c == 1)
        v = pts[(size_t)(b * n + j) * 3 + 1] - cy;
      else if (c == 2)
        v = pts[(size_t)(b * n + j) * 3 + 2] - cz;
      else if (c < Cin0)
        v = fin[(size_t)(b * n + j) * Cprev + (c - 3)];
      bufA[k * CP0 + c] = v;
    }
    __syncthreads();
    mlp_layer16(bufA, bufB, CP0, CP1, L0, tid);
    __syncthreads();
    mlp_layer16(bufB, bufA, CP1, CP2, L1, tid);
    __syncthreads();
    mlp_layer16_pool(bufA, pool, CP2, L2, tid);
    __syncthreads();
  }

  for (int o = tid; o < L2.Cout; o += 256)
    fout[(size_t)g * L2.Cout + o] = __int_as_float(pool[o]);
}

// ---------------------------------------------------------------------------
// Final outputs: [B,16,3] -> [B,3,16] then [B,16,512] -> [B,512,16], concat.
// ---------------------------------------------------------------------------
__global__ void k_writeout(const float* __restrict__ ctr3,
                           const float* __restrict__ np3,
                           float* __restrict__ out) {
  int i = blockIdx.x * blockDim.x + threadIdx.x;
  if (i < 384) {
    int b = i / 48, rr = i - b * 48;
    int c = rr / 16, s = rr - c * 16;
    out[i] = ctr3[(size_t)(b * 16 + s) * 3 + c];
  } else if (i < 65920) {
    int t = i - 384;
    int b = t / 8192, rr = t - b * 8192;
    int o = rr / 16, s = rr - o * 16;
    out[i] = np3[(size_t)(b * 16 + s) * 512 + o];
  }
}

// ---------------------------------------------------------------------------
extern "C" void kernel_launch(void* const* d_in, const int* in_sizes, int n_in,
                              void* d_out, int out_size, void* d_ws,
                              size_t ws_size, hipStream_t stream) {
  (void)out_size;
  (void)ws_size;
  const float* xyz;
  const float* P[72];
  if (n_in >= 73 && in_sizes[0] == BATCH * 6 * NPTS) {
    xyz = (const float*)d_in[0];
    for (int i = 0; i < 72; ++i) P[i] = (const float*)d_in[1 + i];
  } else {  // params flattened first, xyz last
    for (int i = 0; i < 72; ++i) P[i] = (const float*)d_in[i];
    xyz = (const float*)d_in[n_in - 1];
  }

  float* f = (float*)d_ws;
  float* pts0 = f;   f += (size_t)BATCH * NPTS * 3;
  float* feat0 = f;  f += (size_t)BATCH * NPTS * 6;
  float* nx0 = f;    f += (size_t)BATCH * 1024 * 3;
  float* nx1 = f;    f += (size_t)BATCH * 256 * 3;
  float* nx2 = f;    f += (size_t)BATCH * 64 * 3;
  float* nx3 = f;    f += (size_t)BATCH * 16 * 3;
  float* np0 = f;    f += (size_t)BATCH * 1024 * 64;
  float* np1 = f;    f += (size_t)BATCH * 256 * 128;
  float* np2 = f;    f += (size_t)BATCH * 64 * 256;
  float* np3 = f;    f += (size_t)BATCH * 16 * 512;
  int* fidx = (int*)f;
  f += (size_t)BATCH * 1024;  // fps index scratch (ints)

  static const int LCIN[12] = {9,  32,  32,  67,  64,  64,
                               131, 128, 128, 259, 256, 256};
  static const int LCOUT[12] = {32, 32,  64,  64,  64,  128,
                                128, 128, 256, 256, 256, 512};
  float* wp[12];
  int kp[12];
  for (int l = 0; l < 12; ++l) {
    int Kp = (LCIN[l] + 3) & ~3;
    kp[l] = Kp;
    wp[l] = f;
    f += (size_t)LCOUT[l] * Kp;
    int tot = LCOUT[l] * Kp;
    k_pad_w<<<(tot + 255) / 256, 256, 0, stream>>>(P[l * 6 + 0], wp[l],
                                                   LCIN[l], Kp, LCOUT[l]);
  }

  auto mkL = [&](int l) {
    LayerP L;
    L.W = wp[l];
    L.bb = P[l * 6 + 1];
    L.g = P[l * 6 + 2];
    L.be = P[l * 6 + 3];
    L.mu = P[l * 6 + 4];
    L.var = P[l * 6 + 5];
    L.Kp = kp[l];
    L.Cout = LCOUT[l];
    return L;
  };

  dim3 blk(256);
  k_transpose_in<<<(BATCH * NPTS + 255) / 256, blk, 0, stream>>>(xyz, pts0,
                                                                 feat0);
  // stage 0: 4096 -> 1024 centers, r=0.1, MLP 9->32->32->64
  k_fps<<<BATCH, blk, 0, stream>>>(pts0, NPTS, 1024, fidx);
  k_gather_centers<<<(BATCH * 1024 + 255) / 256, blk, 0, stream>>>(
      pts0, fidx, nx0, NPTS, 1024);
  k_sa_stage<36, 36><<<BATCH * 1024, blk, 0, stream>>>(
      pts0, feat0, nx0, np0, NPTS, 1024, 6, 0.01f, mkL(0), mkL(1), mkL(2));
  // stage 1: 1024 -> 256, r=0.2, MLP 67->64->64->128
  k_fps<<<BATCH, blk, 0, stream>>>(nx0, 1024, 256, fidx);
  k_gather_centers<<<(BATCH * 256 + 255) / 256, blk, 0, stream>>>(
      nx0, fidx, nx1, 1024, 256);
  k_sa_stage<72, 68><<<BATCH * 256, blk, 0, stream>>>(
      nx0, np0, nx1, np1, 1024, 256, 64, 0.04f, mkL(3), mkL(4), mkL(5));
  // stage 2: 256 -> 64, r=0.4, MLP 131->128->128->256
  k_fps<<<BATCH, blk, 0, stream>>>(nx1, 256, 64, fidx);
  k_gather_centers<<<(BATCH * 64 + 255) / 256, blk, 0, stream>>>(
      nx1, fidx, nx2, 256, 64);
  k_sa_stage<136, 132><<<BATCH * 64, blk, 0, stream>>>(
      nx1, np1, nx2, np2, 256, 64, 128, 0.16f, mkL(6), mkL(7), mkL(8));
  // stage 3: 64 -> 16, r=0.8, MLP 259->256->256->512
  k_fps<<<BATCH, blk, 0, stream>>>(nx2, 64, 16, fidx);
  k_gather_centers<<<(BATCH * 16 + 255) / 256, blk, 0, stream>>>(
      nx2, fidx, nx3, 64, 16);
  k_sa_stage<264, 260><<<BATCH * 16, blk, 0, stream>>>(
      nx2, np2, nx3, np3, 64, 16, 256, 0.64f, mkL(9), mkL(10), mkL(11));

  k_writeout<<<(65920 + 255) / 256, blk, 0, stream>>>(nx3, np3,
                                                      (float*)d_out);
}